// RelativePositionEmbedding_88115549045052
// MI455X (gfx1250) — compile-verified
//
#include <hip/hip_runtime.h>

typedef float v2f __attribute__((ext_vector_type(2)));
typedef float v4f __attribute__((ext_vector_type(4)));
typedef float v8f __attribute__((ext_vector_type(8)));

#define MAX_POS 1024
#define S_LEN   1024
#define EMB     1024
#define HEADS   16
#define DHEAD   64
#define BATCH   2
#define M1_PAD  2048   // padded row count for emb2 (2047 real rows)

// ---------------------------------------------------------------------------
// GEMM1: emb2[m][e] = sum_k pos[clamp(MAX_POS-S+m)][k] * W[k][e]
// 64x64 tile per block, K chunks of 32, 128 threads = 4 waves (2x2),
// each wave computes 32x32 via 2x2 V_WMMA_F32_16X16X4_F32 tiles.
// ---------------------------------------------------------------------------
#define LDA1 36   // LDS row stride (floats): 36*4=144B -> float4-aligned rows
#define LDB1 36

__global__ __launch_bounds__(128)
void gemm1_kernel(const float* __restrict__ pos, const float* __restrict__ W,
                  float* __restrict__ emb2)
{
    __shared__ float Als[64 * LDA1];   // [m][k]   64 x 32
    __shared__ float Bls[64 * LDB1];   // [n][k]   (transposed on load) 64 x 32

    const int t    = threadIdx.x;
    const int lane = t & 31;
    const int wave = t >> 5;
    const int wm   = wave >> 1;   // wave's M half
    const int wn   = wave & 1;    // wave's N half

    const int Mbase = blockIdx.x * 64;
    const int Nbase = blockIdx.y * 64;

    v8f acc[2][2] = {};

    for (int kc = 0; kc < EMB; kc += 32) {
        // --- A tile: 64 rows (gather-clamped) x 32 k, float4 coalesced ---
        #pragma unroll
        for (int it = 0; it < 4; ++it) {
            int idx = t + it * 128;              // 512 float4
            int row = idx >> 3;                  // 8 float4 per 32-float row
            int col = (idx & 7) << 2;
            int gidx = Mbase + row + (MAX_POS - S_LEN);
            gidx = gidx < 0 ? 0 : gidx;
            gidx = gidx > (2 * MAX_POS - 2) ? (2 * MAX_POS - 2) : gidx;
            v4f v = *(const v4f*)(pos + (size_t)gidx * EMB + kc + col);
            *(v4f*)(&Als[row * LDA1 + col]) = v;
        }
        // --- B tile: 32 k x 64 n, coalesced read, transposed LDS store ---
        #pragma unroll
        for (int it = 0; it < 16; ++it) {
            int idx = t + it * 128;              // 2048 scalars
            int k = idx >> 6;
            int n = idx & 63;
            Bls[n * LDB1 + k] = W[(size_t)(kc + k) * EMB + Nbase + n];
        }
        __syncthreads();

        #pragma unroll
        for (int ks = 0; ks < 32; ks += 4) {
            v2f afrag[2], bfrag[2];
            #pragma unroll
            for (int tm = 0; tm < 2; ++tm) {
                int m = wm * 32 + tm * 16 + (lane & 15);
                afrag[tm] = *(const v2f*)(&Als[m * LDA1 + ks + 2 * (lane >> 4)]);
            }
            #pragma unroll
            for (int tn = 0; tn < 2; ++tn) {
                int n = wn * 32 + tn * 16 + (lane & 15);
                bfrag[tn] = *(const v2f*)(&Bls[n * LDB1 + ks + 2 * (lane >> 4)]);
            }
            #pragma unroll
            for (int tm = 0; tm < 2; ++tm)
                #pragma unroll
                for (int tn = 0; tn < 2; ++tn)
                    acc[tm][tn] = __builtin_amdgcn_wmma_f32_16x16x4_f32(
                        false, afrag[tm], false, bfrag[tn],
                        (short)0, acc[tm][tn], false, false);
        }
        __syncthreads();
    }

    // C/D layout: VGPR v -> row v + 8*(lane/16), col = lane%16
    #pragma unroll
    for (int tm = 0; tm < 2; ++tm)
        #pragma unroll
        for (int tn = 0; tn < 2; ++tn) {
            int mrow0 = Mbase + wm * 32 + tm * 16 + 8 * (lane >> 4);
            int ncol  = Nbase + wn * 32 + tn * 16 + (lane & 15);
            #pragma unroll
            for (int v = 0; v < 8; ++v)
                emb2[(size_t)(mrow0 + v) * EMB + ncol] = acc[tm][tn][v];
        }
}

// ---------------------------------------------------------------------------
// GEMM2 (+ fused relative shift): per head h,
//   score[h][n][m] = sum_d q[h][n][d] * emb2[m][h*64+d]
// epilogue scatters score[h][n=b*S+s][m] to out[((h*B+b)*S+s)*S + j],
// j = m - (S-1) + s, skipping j outside [0,S). Score never hits HBM.
// ---------------------------------------------------------------------------
#define LDA2 68   // 68*4 = 272B rows -> float4-aligned
#define LDB2 68

__global__ __launch_bounds__(128)
void gemm2_kernel(const float* __restrict__ query, const float* __restrict__ emb2,
                  float* __restrict__ out)
{
    __shared__ float Als[64 * LDA2];   // q rows   [n][d]  64 x 64
    __shared__ float Bls[64 * LDB2];   // emb rows [m][d]  64 x 64

    const int t    = threadIdx.x;
    const int lane = t & 31;
    const int wave = t >> 5;
    const int wm   = wave >> 1;   // n half
    const int wn   = wave & 1;    // m half

    const int h     = blockIdx.z;
    const int Nbase = blockIdx.x * 64;   // n in [0, B*S)
    const int Mbase = blockIdx.y * 64;   // m in [0, 2048) padded

    // load q tile (coalesced float4)
    #pragma unroll
    for (int it = 0; it < 8; ++it) {
        int idx = t + it * 128;
        int row = idx >> 4;
        int col = (idx & 15) << 2;
        v4f v = *(const v4f*)(query +
                 ((size_t)h * (BATCH * S_LEN) + Nbase + row) * DHEAD + col);
        *(v4f*)(&Als[row * LDA2 + col]) = v;
    }
    // load emb tile (coalesced float4)
    #pragma unroll
    for (int it = 0; it < 8; ++it) {
        int idx = t + it * 128;
        int row = idx >> 4;
        int col = (idx & 15) << 2;
        v4f v = *(const v4f*)(emb2 + (size_t)(Mbase + row) * EMB + h * DHEAD + col);
        *(v4f*)(&Bls[row * LDB2 + col]) = v;
    }
    __syncthreads();

    v8f acc[2][2] = {};
    #pragma unroll
    for (int ks = 0; ks < DHEAD; ks += 4) {
        v2f afrag[2], bfrag[2];
        #pragma unroll
        for (int tm = 0; tm < 2; ++tm) {
            int n = wm * 32 + tm * 16 + (lane & 15);
            afrag[tm] = *(const v2f*)(&Als[n * LDA2 + ks + 2 * (lane >> 4)]);
        }
        #pragma unroll
        for (int tn = 0; tn < 2; ++tn) {
            int m = wn * 32 + tn * 16 + (lane & 15);
            bfrag[tn] = *(const v2f*)(&Bls[m * LDB2 + ks + 2 * (lane >> 4)]);
        }
        #pragma unroll
        for (int tm = 0; tm < 2; ++tm)
            #pragma unroll
            for (int tn = 0; tn < 2; ++tn)
                acc[tm][tn] = __builtin_amdgcn_wmma_f32_16x16x4_f32(
                    false, afrag[tm], false, bfrag[tn],
                    (short)0, acc[tm][tn], false, false);
    }

    // fused relative-shift scatter epilogue
    #pragma unroll
    for (int tm = 0; tm < 2; ++tm)
        #pragma unroll
        for (int tn = 0; tn < 2; ++tn) {
            int nrow0 = Nbase + wm * 32 + tm * 16 + 8 * (lane >> 4);
            int mcol  = Mbase + wn * 32 + tn * 16 + (lane & 15);
            #pragma unroll
            for (int v = 0; v < 8; ++v) {
                int n = nrow0 + v;
                int b = n >> 10;            // n / S
                int s = n & (S_LEN - 1);    // n % S
                int j = mcol - (S_LEN - 1) + s;
                if ((unsigned)j < (unsigned)S_LEN) {
                    size_t o = (((size_t)h * BATCH + b) * S_LEN + s) * (size_t)S_LEN + j;
                    out[o] = acc[tm][tn][v];
                }
            }
        }
}

extern "C" void kernel_launch(void* const* d_in, const int* in_sizes, int n_in,
                              void* d_out, int out_size, void* d_ws, size_t ws_size,
                              hipStream_t stream) {
    const float* query = (const float*)d_in[0];   // [2,16,1024,64]
    const float* pos   = (const float*)d_in[1];   // [2047,1024]
    const float* W     = (const float*)d_in[2];   // [1024,1024]
    float* out  = (float*)d_out;                  // [2,16,1024,1024]
    float* emb2 = (float*)d_ws;                   // [2048,1024] = 8 MB scratch

    dim3 g1(M1_PAD / 64, EMB / 64);               // 32 x 16
    gemm1_kernel<<<g1, 128, 0, stream>>>(pos, W, emb2);

    dim3 g2((BATCH * S_LEN) / 64, M1_PAD / 64, HEADS);  // 32 x 32 x 16
    gemm2_kernel<<<g2, 128, 0, stream>>>(query, emb2, out);
}